// MultiObjectONet_49177375539879
// MI455X (gfx1250) — compile-verified
//
#include <hip/hip_runtime.h>
#include <stdint.h>

#define BS    16
#define NP    500000
#define NOBJ  16

// presence pass: 50 blocks/row -> 10000 tags/block (start offset 16B-aligned)
#define PRES_BPR  50
#define PRES_CH   (NP / PRES_BPR)        // 10000
#define PRES_TILE 1024                   // tags per tile (256 thr * 4 tags)
#define PRES_FULL (PRES_CH / PRES_TILE)  // 9 full tiles, 784 tail

// main pass: 50 blocks/row -> exactly 10000 points/block (2500 4-point groups)
#define MAIN_BPR 50
#define MAIN_CH  (NP / MAIN_BPR)         // 10000 (multiple of 4; 16B-aligned starts)

// ---- order-preserving float <-> uint mapping (no NaNs in this workload) ----
__device__ __forceinline__ unsigned fmap(float f) {
  unsigned u = __float_as_uint(f);
  return (u & 0x80000000u) ? ~u : (u | 0x80000000u);
}
__device__ __forceinline__ float funmap(unsigned u) {
  return (u & 0x80000000u) ? __uint_as_float(u & 0x7fffffffu) : __uint_as_float(~u);
}

// ---- CDNA5 async global->LDS (VGLOBAL async encoding, GV mode, ASYNCcnt) ----
__device__ __forceinline__ void async_ld_b128(unsigned lds_byte_off, const void* gaddr) {
  asm volatile("global_load_async_to_lds_b128 %0, %1, off"
               :: "v"(lds_byte_off), "v"(gaddr) : "memory");
}
__device__ __forceinline__ void wait_async_le1() {
  asm volatile("s_wait_asynccnt 1" ::: "memory");
}
__device__ __forceinline__ void wait_async_0() {
  asm volatile("s_wait_asynccnt 0" ::: "memory");
}

// ---- workspace init (d_ws is poisoned; must init every call) ----
__global__ void k_init(unsigned* __restrict__ ws) {
  for (int i = threadIdx.x; i < 16 + 768 + 768; i += 256) {
    unsigned v;
    if (i < 16)            v = 0u;           // present masks
    else if (i < 16 + 768) v = 0xFFFFFFFFu;  // minU (mapped +inf)
    else                   v = 0u;           // maxU (mapped -inf)
    ws[i] = v;
  }
}

// ---- pass A: per-row tag presence bitmask, tags streamed via async->LDS ----
__global__ void __launch_bounds__(256) k_presence(const int* __restrict__ tags,
                                                  unsigned* __restrict__ present) {
  __shared__ __align__(16) int stage[2][PRES_TILE];   // 8KB double buffer
  const int tid = threadIdx.x;
  const int row = blockIdx.y;
  const int start = blockIdx.x * PRES_CH;             // 16B-aligned element offset
  const int* base = tags + (size_t)row * NP + start;
  unsigned mask = 0u;

  // low 32 bits of a generic LDS pointer == wave-relative LDS byte offset
  const unsigned slot0 = (unsigned)(size_t)(&stage[0][tid * 4]);
  const unsigned slot1 = (unsigned)(size_t)(&stage[1][tid * 4]);

  async_ld_b128(slot0, base + tid * 4);               // tile 0 -> buf 0
  for (int it = 0; it < PRES_FULL; ++it) {
    if (it + 1 < PRES_FULL) {
      // tile it+1 -> buffer (it+1)&1
      async_ld_b128((it & 1) ? slot0 : slot1, base + (it + 1) * PRES_TILE + tid * 4);
      wait_async_le1();                               // tile it complete
    } else {
      wait_async_0();
    }
    // each lane consumes only its own staged slot -> no cross-wave sync needed
    const int4 v = *reinterpret_cast<const int4*>(&stage[it & 1][tid * 4]);
    mask |= (1u << v.x) | (1u << v.y) | (1u << v.z) | (1u << v.w);
  }
  for (int i = PRES_FULL * PRES_TILE + tid; i < PRES_CH; i += 256)
    mask |= 1u << base[i];

  __shared__ unsigned bm;
  if (tid == 0) bm = 0u;
  __syncthreads();
  atomicOr(&bm, mask);
  __syncthreads();
  if (tid == 0) atomicOr(&present[row], bm);
}

// ---- pass B: relabel + obj_batch output + segmented min/max via LDS atomics ----
__global__ void __launch_bounds__(256) k_main(const float* __restrict__ pc,
                                              const int* __restrict__ tags,
                                              const unsigned* __restrict__ present,
                                              unsigned* __restrict__ minU,
                                              unsigned* __restrict__ maxU,
                                              float* __restrict__ objOut) {
  // 32 per-8-lane-group tables, 96 entries each (obj*6: min xyz, max xyz),
  // stride 97 dwords to decorrelate LDS banks across groups.  ~12.4KB.
  __shared__ unsigned tbl[32 * 97];
  const int tid = threadIdx.x;
  const int row = blockIdx.y;

  for (int i = tid; i < 32 * 97; i += 256) {
    const int m = i % 97;
    tbl[i] = (m < 96 && (m % 6) >= 3) ? 0u : 0xFFFFFFFFu;
  }
  __syncthreads();

  const unsigned mask = present[row];
  const size_t rowbase = (size_t)row * NP;
  const float rowofs = (float)(row * NOBJ);
  unsigned* mytbl = &tbl[(tid >> 3) * 97];

  // 16B-aligned vector views of this row (row offsets: 2MB / 6MB / 2MB)
  const int4*   tagv = (const int4*)(tags + rowbase);
  const float4* pcv  = (const float4*)(pc + rowbase * 3);
  float4*       outv = (float4*)(objOut + rowbase);

  const int g0   = (blockIdx.x * MAIN_CH) >> 2;        // first 4-point group
  const int gEnd = g0 + (MAIN_CH >> 2);                // exactly 2500 groups/block

  for (int g = g0 + tid; g < gEnd; g += 256) {
    const int4   t = tagv[g];                          // global_load_b128
    const float4 a = pcv[3 * g + 0];                   // 3x global_load_b128
    const float4 b = pcv[3 * g + 1];
    const float4 c = pcv[3 * g + 2];
    // points: p0=(a.x,a.y,a.z) p1=(a.w,b.x,b.y) p2=(b.z,b.w,c.x) p3=(c.y,c.z,c.w)
    const int r0 = __popc(mask & ((2u << t.x) - 1u)) - 1;
    const int r1 = __popc(mask & ((2u << t.y) - 1u)) - 1;
    const int r2 = __popc(mask & ((2u << t.z) - 1u)) - 1;
    const int r3 = __popc(mask & ((2u << t.w) - 1u)) - 1;
    float4 o;
    o.x = (float)r0 + rowofs; o.y = (float)r1 + rowofs;
    o.z = (float)r2 + rowofs; o.w = (float)r3 + rowofs;
    outv[g] = o;                                       // global_store_b128

    unsigned* e0 = mytbl + r0 * 6;
    atomicMin(e0 + 0, fmap(a.x)); atomicMin(e0 + 1, fmap(a.y)); atomicMin(e0 + 2, fmap(a.z));
    atomicMax(e0 + 3, fmap(a.x)); atomicMax(e0 + 4, fmap(a.y)); atomicMax(e0 + 5, fmap(a.z));
    unsigned* e1 = mytbl + r1 * 6;
    atomicMin(e1 + 0, fmap(a.w)); atomicMin(e1 + 1, fmap(b.x)); atomicMin(e1 + 2, fmap(b.y));
    atomicMax(e1 + 3, fmap(a.w)); atomicMax(e1 + 4, fmap(b.x)); atomicMax(e1 + 5, fmap(b.y));
    unsigned* e2 = mytbl + r2 * 6;
    atomicMin(e2 + 0, fmap(b.z)); atomicMin(e2 + 1, fmap(b.w)); atomicMin(e2 + 2, fmap(c.x));
    atomicMax(e2 + 3, fmap(b.z)); atomicMax(e2 + 4, fmap(b.w)); atomicMax(e2 + 5, fmap(c.x));
    unsigned* e3 = mytbl + r3 * 6;
    atomicMin(e3 + 0, fmap(c.y)); atomicMin(e3 + 1, fmap(c.z)); atomicMin(e3 + 2, fmap(c.w));
    atomicMax(e3 + 3, fmap(c.y)); atomicMax(e3 + 4, fmap(c.z)); atomicMax(e3 + 5, fmap(c.w));
  }
  __syncthreads();

  // merge 32 group-tables, then one global atomic per (segment, value)
  for (int v = tid; v < 96; v += 256) {
    const bool isMin = (v % 6) < 3;
    unsigned u = tbl[v];
#pragma unroll
    for (int g = 1; g < 32; ++g) {
      const unsigned u2 = tbl[g * 97 + v];
      u = isMin ? min(u, u2) : max(u, u2);
    }
    const int rank = v / 6;
    const int c = isMin ? (v % 6) : (v % 6) - 3;
    const int gidx = (row * NOBJ + rank) * 3 + c;
    if (isMin) atomicMin(&minU[gidx], u);
    else       atomicMax(&maxU[gidx], u);
  }
}

// ---- pass C: bbox interleave + faithful torch-style barycenter ----
__global__ void k_final(const unsigned* __restrict__ present,
                        const unsigned* __restrict__ minU,
                        const unsigned* __restrict__ maxU,
                        float* __restrict__ out) {
  __shared__ float bb[BS][2 * NOBJ][3];
  const int tid = threadIdx.x;        // 256 threads = (b,o) pairs
  const int b = tid >> 4;
  const int o = tid & 15;
  const int K = __popc(present[b]);   // nonempty segments == o < K
  const int s = b * NOBJ + o;
  for (int c = 0; c < 3; ++c) {
    float mn = 0.f, mx = 0.f;
    if (o < K) { mn = funmap(minU[s * 3 + c]); mx = funmap(maxU[s * 3 + c]); }
    bb[b][2 * o][c] = mn;
    bb[b][2 * o + 1][c] = mx;
  }
  __syncthreads();
  for (int c = 0; c < 3; ++c) {
    out[b * 144 + (2 * o) * 3 + c]     = bb[b][2 * o][c];
    out[b * 144 + (2 * o + 1) * 3 + c] = bb[b][2 * o + 1][c];
    // reshape(bs,2,n_obj,3).mean(axis=1): bary row o = (bbox_row[o]+bbox_row[16+o])/2
    out[b * 144 + (32 + o) * 3 + c]    = 0.5f * (bb[b][o][c] + bb[b][16 + o][c]);
  }
}

extern "C" void kernel_launch(void* const* d_in, const int* in_sizes, int n_in,
                              void* d_out, int out_size, void* d_ws, size_t ws_size,
                              hipStream_t stream) {
  const float* pc   = (const float*)d_in[0];
  const int*   tags = (const int*)d_in[1];
  float* out = (float*)d_out;

  // ws layout (uints): [0,16) present | [16,784) minU | [784,1552) maxU  (6208 B)
  unsigned* ws      = (unsigned*)d_ws;
  unsigned* present = ws;
  unsigned* minU    = ws + 16;
  unsigned* maxU    = ws + 16 + 768;

  k_init<<<1, 256, 0, stream>>>(ws);
  k_presence<<<dim3(PRES_BPR, BS), 256, 0, stream>>>(tags, present);
  k_main<<<dim3(MAIN_BPR, BS), 256, 0, stream>>>(pc, tags, present, minU, maxU,
                                                 out + BS * 3 * NOBJ * 3);
  k_final<<<1, 256, 0, stream>>>(present, minU, maxU, out);
}